// ProbAttention_26371099197992
// MI455X (gfx1250) — compile-verified
//
#include <hip/hip_runtime.h>
#include <hip/hip_bf16.h>
#include <math.h>

#define B_ 8
#define L_ 4096
#define H_ 8
#define D_ 64
#define U_ 45          // sample count == n_top == 45
#define NEG_BIG -3e38f

typedef __attribute__((ext_vector_type(2))) float v2f;
typedef __attribute__((ext_vector_type(8))) float v8f;

// ---------------------------------------------------------------------------
// Kernel 1: sparsity measure  M[b,h,l] = max_s(Q.K_sample) - sum_s(...)/L
// one wave32 per (b,h,l); lanes split D=64 as float2; shuffle-reduce dots.
// ---------------------------------------------------------------------------
__global__ void k_meas(const float* __restrict__ Q, const float* __restrict__ K,
                       const int* __restrict__ idxs, float* __restrict__ Mbuf) {
  int wave = (blockIdx.x * blockDim.x + threadIdx.x) >> 5;
  int lane = threadIdx.x & 31;
  int l = wave % L_;
  int h = (wave / L_) % H_;
  int b = wave / (L_ * H_);
  const float* qrow = Q + (((size_t)b * L_ + l) * H_ + h) * D_;
  float2 q = *(const float2*)(qrow + lane * 2);
  float maxv = NEG_BIG, sumv = 0.f;
  for (int s = 0; s < U_; ++s) {
    int ki = idxs[l * U_ + s];
    const float* krow = K + (((size_t)b * L_ + ki) * H_ + h) * D_;
    float2 kv = *(const float2*)(krow + lane * 2);
    float p = q.x * kv.x + q.y * kv.y;
    for (int off = 16; off; off >>= 1) p += __shfl_xor(p, off, 32);
    maxv = fmaxf(maxv, p);
    sumv += p;
  }
  if (lane == 0) Mbuf[wave] = maxv - sumv * (1.0f / (float)L_);
}

// ---------------------------------------------------------------------------
// Kernel 2: top-45 indices of M per (b,h); iterative argmax in LDS.
// ---------------------------------------------------------------------------
__global__ void k_topk(const float* __restrict__ Mbuf, int* __restrict__ Mtop) {
  __shared__ float sm[L_];
  __shared__ float rv[256];
  __shared__ int   ri[256];
  int bh = blockIdx.x;
  const float* row = Mbuf + (size_t)bh * L_;
  for (int i = threadIdx.x; i < L_; i += 256) sm[i] = row[i];
  __syncthreads();
  for (int t = 0; t < U_; ++t) {
    float best = NEG_BIG; int bi = 0;
    for (int i = threadIdx.x; i < L_; i += 256) {
      float v = sm[i];
      if (v > best) { best = v; bi = i; }
    }
    rv[threadIdx.x] = best; ri[threadIdx.x] = bi;
    __syncthreads();
    for (int s = 128; s; s >>= 1) {
      if (threadIdx.x < s) {
        float o = rv[threadIdx.x + s]; int oi = ri[threadIdx.x + s];
        float m = rv[threadIdx.x];
        if (o > m || (o == m && oi < ri[threadIdx.x])) {
          rv[threadIdx.x] = o; ri[threadIdx.x] = oi;
        }
      }
      __syncthreads();
    }
    if (threadIdx.x == 0) { Mtop[bh * U_ + t] = ri[0]; sm[ri[0]] = NEG_BIG; }
    __syncthreads();
  }
}

// ---------------------------------------------------------------------------
// Kernels 3a/3b: context = cumsum(V) over L, chunked scan (16 chunks of 256).
// Output layout is (B,H,L,D); input V layout is (B,L,H,D).
// ---------------------------------------------------------------------------
__global__ void k_csum_part(const float* __restrict__ V, float* __restrict__ Csum) {
  int blk = blockIdx.x;                 // B*H*16
  int c = blk % 16;
  int h = (blk / 16) % H_;
  int b = blk / (16 * H_);
  int d = threadIdx.x;                  // 64
  float s = 0.f;
  int l0 = c * 256;
  for (int i = 0; i < 256; ++i)
    s += V[(((size_t)b * L_ + l0 + i) * H_ + h) * D_ + d];
  Csum[((size_t)(b * H_ + h) * 16 + c) * D_ + d] = s;
}

__global__ void k_csum_write(const float* __restrict__ V, const float* __restrict__ Csum,
                             float* __restrict__ out) {
  int blk = blockIdx.x;
  int c = blk % 16;
  int h = (blk / 16) % H_;
  int b = blk / (16 * H_);
  int d = threadIdx.x;
  float run = 0.f;
  for (int cc = 0; cc < c; ++cc)
    run += Csum[((size_t)(b * H_ + h) * 16 + cc) * D_ + d];
  int l0 = c * 256;
  for (int i = 0; i < 256; ++i) {
    int l = l0 + i;
    run += V[(((size_t)b * L_ + l) * H_ + h) * D_ + d];
    out[(((size_t)(b * H_ + h)) * L_ + l) * D_ + d] = run;
  }
}

// ---------------------------------------------------------------------------
// Kernel 4: dense attention for the 45 selected queries using fp32 WMMA.
// grid = B*H*3 (one 16-row M-tile per block), block = 1 wave32.
// Pass A: scores tile = WMMA(Q_red, K^T), masked online softmax stats.
// Pass B: recompute, P = exp(s-m)/l via LDS relayout, acc += WMMA(P, V).
// ---------------------------------------------------------------------------
__global__ void __launch_bounds__(32)
k_attn(const float* __restrict__ Q, const float* __restrict__ K,
       const float* __restrict__ V, const int* __restrict__ Mtop,
       float* __restrict__ out) {
  int blk = blockIdx.x;
  int mt = blk % 3;
  int h = (blk / 3) % H_;
  int b = blk / (3 * H_);
  int lane = threadIdx.x;

  __shared__ int   s_rows[16];
  __shared__ float s_p[16 * 16];

  int u0 = mt * 16;
  if (lane < 16) {
    int u = u0 + lane;
    s_rows[lane] = Mtop[(b * H_ + h) * U_ + ((u < U_) ? u : (U_ - 1))];
  }
  __syncthreads();

  int half = lane >> 4;       // 0 or 1
  int lo   = lane & 15;
  int rowAdd = half * 8;      // C-tile rows r+rowAdd per lane half

  int bounds[8];
  for (int r = 0; r < 8; ++r) bounds[r] = s_rows[r + rowAdd];

  // A fragments of Q_reduce (16x64 fp32): 16 k-steps of 4
  v2f afrag[16];
  {
    const float* qbase = Q + (((size_t)b * L_ + s_rows[lo]) * H_ + h) * D_;
    for (int ks = 0; ks < 16; ++ks) {
      float2 t = *(const float2*)(qbase + ks * 4 + half * 2);
      v2f a; a.x = t.x; a.y = t.y;
      afrag[ks] = a;
    }
  }

  const float scale = 0.125f;   // 1/sqrt(64)
  const int   rstride = H_ * D_;
  const float* Kbase = K + ((size_t)b * L_) * rstride + h * D_;
  const float* Vbase = V + ((size_t)b * L_) * rstride + h * D_;

  float mrun[8], srun[8];
  for (int r = 0; r < 8; ++r) { mrun[r] = NEG_BIG; srun[r] = 0.f; }

  // ---------------- Pass A: softmax statistics ----------------
  for (int nt = 0; nt < L_ / 16; ++nt) {
    v8f c = {0.f, 0.f, 0.f, 0.f, 0.f, 0.f, 0.f, 0.f};
    const float* krow = Kbase + (size_t)(nt * 16 + lo) * rstride;
    for (int ks = 0; ks < 16; ++ks) {
      float2 t = *(const float2*)(krow + ks * 4 + half * 2);
      v2f bf; bf.x = t.x; bf.y = t.y;
      c = __builtin_amdgcn_wmma_f32_16x16x4_f32(false, afrag[ks], false, bf,
                                                (short)0, c, false, false);
    }
    int col = nt * 16 + lo;
#pragma unroll
    for (int r = 0; r < 8; ++r) {
      if (col <= bounds[r]) {             // masked entries contribute exactly 0
        float sc = c[r] * scale;
        float nm = fmaxf(mrun[r], sc);
        srun[r] = srun[r] * __expf(mrun[r] - nm) + __expf(sc - nm);
        mrun[r] = nm;
      }
    }
  }
  // combine per-lane partials across the 16 lanes of each half-wave
#pragma unroll
  for (int r = 0; r < 8; ++r) {
    float m = mrun[r], s = srun[r];
    for (int off = 8; off; off >>= 1) {
      float mo = __shfl_xor(m, off, 32);
      float so = __shfl_xor(s, off, 32);
      float nm = fmaxf(m, mo);
      s = s * __expf(m - nm) + so * __expf(mo - nm);
      m = nm;
    }
    mrun[r] = m; srun[r] = s;
  }

  // ---------------- Pass B: P @ V ----------------
  v8f acc[4];
  for (int vt = 0; vt < 4; ++vt)
    for (int i = 0; i < 8; ++i) acc[vt][i] = 0.f;

  for (int nt = 0; nt < L_ / 16; ++nt) {
    v8f c = {0.f, 0.f, 0.f, 0.f, 0.f, 0.f, 0.f, 0.f};
    const float* krow = Kbase + (size_t)(nt * 16 + lo) * rstride;
    for (int ks = 0; ks < 16; ++ks) {
      float2 t = *(const float2*)(krow + ks * 4 + half * 2);
      v2f bf; bf.x = t.x; bf.y = t.y;
      c = __builtin_amdgcn_wmma_f32_16x16x4_f32(false, afrag[ks], false, bf,
                                                (short)0, c, false, false);
    }
    int col = nt * 16 + lo;
    float p[8];
#pragma unroll
    for (int r = 0; r < 8; ++r)
      p[r] = (col <= bounds[r]) ? __expf(c[r] * scale - mrun[r]) / srun[r] : 0.f;

    __syncthreads();
#pragma unroll
    for (int r = 0; r < 8; ++r) s_p[(r + rowAdd) * 16 + lo] = p[r];
    __syncthreads();

    // P tile (16x16) re-read in A-matrix layout: 4 k-steps of 4
    v2f pfrag[4];
#pragma unroll
    for (int ks = 0; ks < 4; ++ks) {
      int pk = ks * 4 + half * 2;
      v2f a; a.x = s_p[lo * 16 + pk]; a.y = s_p[lo * 16 + pk + 1];
      pfrag[ks] = a;
    }

    const float* vrow0 = Vbase + (size_t)(nt * 16) * rstride;
#pragma unroll
    for (int vt = 0; vt < 4; ++vt) {
#pragma unroll
      for (int ks = 0; ks < 4; ++ks) {
        int kk = ks * 4 + half * 2;
        v2f bf;
        bf.x = vrow0[(size_t)kk       * rstride + vt * 16 + lo];
        bf.y = vrow0[(size_t)(kk + 1) * rstride + vt * 16 + lo];
        acc[vt] = __builtin_amdgcn_wmma_f32_16x16x4_f32(false, pfrag[ks], false, bf,
                                                        (short)0, acc[vt], false, false);
      }
    }
  }

  // scatter updated rows into context (out layout: B,H,L,D)
  for (int vt = 0; vt < 4; ++vt) {
#pragma unroll
    for (int r = 0; r < 8; ++r) {
      int u = u0 + r + rowAdd;
      if (u < U_) {
        int l = s_rows[r + rowAdd];
        out[(((size_t)(b * H_ + h)) * L_ + l) * D_ + vt * 16 + lo] = acc[vt][r];
      }
    }
  }
}

// ---------------------------------------------------------------------------
extern "C" void kernel_launch(void* const* d_in, const int* in_sizes, int n_in,
                              void* d_out, int out_size, void* d_ws, size_t ws_size,
                              hipStream_t stream) {
  const float* Q   = (const float*)d_in[0];
  const float* K   = (const float*)d_in[1];
  const float* V   = (const float*)d_in[2];
  const int*   idx = (const int*)d_in[3];
  float* out = (float*)d_out;

  // workspace carve-up (≈1.3 MB total)
  float* Mbuf = (float*)d_ws;                                   // B*H*L floats
  int*   Mtop = (int*)((char*)d_ws + sizeof(float) * B_ * H_ * L_);  // B*H*45 ints
  float* Csum = (float*)((char*)Mtop + sizeof(int) * B_ * H_ * 64);  // B*H*16*64

  // 1) sparsity measure: one wave per (b,h,l)
  int items = B_ * H_ * L_;
  k_meas<<<items / 8, 256, 0, stream>>>(Q, K, idx, Mbuf);

  // 2) top-45 per (b,h)
  k_topk<<<B_ * H_, 256, 0, stream>>>(Mbuf, Mtop);

  // 3) cumulative sum of V -> out (full coverage of d_out)
  k_csum_part<<<B_ * H_ * 16, D_, 0, stream>>>(V, Csum);
  k_csum_write<<<B_ * H_ * 16, D_, 0, stream>>>(V, Csum, out);

  // 4) WMMA attention for selected queries; overwrites its rows in out
  k_attn<<<B_ * H_ * 3, 32, 0, stream>>>(Q, K, V, Mtop, out);
}